// ExportableGENConv_33174327394958
// MI455X (gfx1250) — compile-verified
//
#include <hip/hip_runtime.h>
#include <hip/hip_bf16.h>

// Sizes fixed by the reference
#define GN_N  32768
#define GN_K  32
#define GN_H  64
#define GN_ED 32
#define GN_H2 128
#define GN_NPB 4   // nodes per block in kernel A (one per wave)

typedef float v2f __attribute__((ext_vector_type(2)));
typedef float v8f __attribute__((ext_vector_type(8)));

static __device__ __forceinline__ v8f wmma_f32(v2f a, v2f b, v8f c) {
  // V_WMMA_F32_16X16X4_F32: D = A(16x4) x B(4x16) + C(16x16)
  return __builtin_amdgcn_wmma_f32_16x16x4_f32(false, a, false, b, (short)0, c,
                                               false, false);
}

// ---------------------------------------------------------------------------
// Kernel A: 4 waves per block, one node per wave; W_edge staged once per block.
// Fused edge message GEMM + neighbor softmax + weighted aggregate + residual.
//   out[i] = x[i] + sum_k alpha_k * msg_k
// ---------------------------------------------------------------------------
__global__ __launch_bounds__(32 * GN_NPB) void k_gnn_fused(
    const float* __restrict__ x,
    const long long* __restrict__ edge_src,   // edge_index row 0
    const long long* __restrict__ nbr,
    const float* __restrict__ edge_attr,
    const float* __restrict__ W_edge,
    float* __restrict__ outbuf)
{
  __shared__ float we[GN_ED * 66];               // W_edge [32][64], shared by block
  __shared__ float ea[GN_NPB][GN_K * 34];        // per-wave edge_attr block
  __shared__ float xs[GN_NPB][GN_K * 68];        // per-wave gathered x[src] rows
  __shared__ int   srcs[GN_NPB][GN_K];
  __shared__ float validf[GN_NPB][GN_K];

  const int tid  = threadIdx.x;
  const int w    = tid >> 5;                     // wave / node slot
  const int lane = tid & 31;
  const int i    = blockIdx.x * GN_NPB + w;      // node id
  const int hlf  = lane >> 4;                    // K-pair half for WMMA fragments
  const int l16  = lane & 15;

  const long long e0 = (long long)i * GN_K;
  srcs[w][lane]   = (int)edge_src[e0 + lane];
  validf[w][lane] = (nbr[e0 + lane] >= 0) ? 1.0f : 0.0f;

  // Stage W_edge once per block (32x64 = 1024 float2 over 128 threads)
  const float2* we_g = reinterpret_cast<const float2*>(W_edge);
#pragma unroll
  for (int t = 0; t < 8; ++t) {
    int li = t * 128 + tid;
    float2 v = we_g[li];
    int row = li >> 5, col = (li & 31) * 2;
    we[row * 66 + col] = v.x; we[row * 66 + col + 1] = v.y;
  }
  // Stage this wave's edge_attr rows (32x32 = 512 float2, fully coalesced)
  const float2* ea_g =
      reinterpret_cast<const float2*>(edge_attr + (size_t)i * GN_K * GN_ED);
#pragma unroll
  for (int t = 0; t < 16; ++t) {
    int li = t * 32 + lane;
    float2 v = ea_g[li];
    int row = li >> 4, col = (li & 15) * 2;
    ea[w][row * 34 + col] = v.x; ea[w][row * 34 + col + 1] = v.y;
  }
  __syncthreads();
  // Gather x[src] rows: one contiguous 256B row per iteration
  for (int r = 0; r < GN_K; ++r) {
    float2 v =
        reinterpret_cast<const float2*>(x + (size_t)srcs[w][r] * GN_H)[lane];
    xs[w][r * 68 + lane * 2]     = v.x;
    xs[w][r * 68 + lane * 2 + 1] = v.y;
  }

  // GEMM: [32 edges x 32] @ [32 x 64]  (2 M-tiles x 4 N-tiles x 8 K-steps)
  v8f acc[2][4] = {};
#pragma unroll
  for (int k0 = 0; k0 < GN_ED; k0 += 4) {
    const int ka = k0 + hlf * 2;
    v2f a0, a1;
    a0.x = ea[w][l16 * 34 + ka];        a0.y = ea[w][l16 * 34 + ka + 1];
    a1.x = ea[w][(16 + l16) * 34 + ka]; a1.y = ea[w][(16 + l16) * 34 + ka + 1];
#pragma unroll
    for (int nt = 0; nt < 4; ++nt) {
      v2f b;
      b.x = we[ka * 66 + nt * 16 + l16];
      b.y = we[(ka + 1) * 66 + nt * 16 + l16];
      acc[0][nt] = wmma_f32(a0, b, acc[0][nt]);
      acc[1][nt] = wmma_f32(a1, b, acc[1][nt]);
    }
  }

  // msg = relu(gemm + x[src]) + 1e-7   (in place in the C-layout accumulators)
#pragma unroll
  for (int mt = 0; mt < 2; ++mt)
#pragma unroll
    for (int nt = 0; nt < 4; ++nt)
#pragma unroll
      for (int r = 0; r < 8; ++r) {
        const int k = mt * 16 + r + hlf * 8;           // edge slot
        float v = acc[mt][nt][r] + xs[w][k * 68 + nt * 16 + l16];
        acc[mt][nt][r] = fmaxf(v, 0.0f) + 1e-7f;
      }

  // Neighbor softmax over the 32 edge slots (per feature column).
  float res[4];
#pragma unroll
  for (int nt = 0; nt < 4; ++nt) {
    float mx = -3.402823466e38f;
#pragma unroll
    for (int mt = 0; mt < 2; ++mt)
#pragma unroll
      for (int r = 0; r < 8; ++r) {
        const int k = mt * 16 + r + hlf * 8;
        mx = fmaxf(mx,
                   (validf[w][k] > 0.0f) ? acc[mt][nt][r] : -3.402823466e38f);
      }
    mx = fmaxf(mx, __shfl_xor(mx, 16, 32));  // merge lane halves (rows 0-7 / 8-15)
    float sex = 0.0f, smex = 0.0f;
#pragma unroll
    for (int mt = 0; mt < 2; ++mt)
#pragma unroll
      for (int r = 0; r < 8; ++r) {
        const int k = mt * 16 + r + hlf * 8;
        const float v = acc[mt][nt][r];
        const float e = __expf(v - mx);
        const bool ok = validf[w][k] > 0.0f;
        sex  += ok ? e : 0.0f;
        smex += ok ? v * e : 0.0f;
      }
    sex  += __shfl_xor(sex, 16, 32);
    smex += __shfl_xor(smex, 16, 32);
    res[nt] = smex / (sex + 1e-16f);
  }
  if (lane < 16) {
#pragma unroll
    for (int nt = 0; nt < 4; ++nt) {
      const size_t idx = (size_t)i * GN_H + nt * 16 + l16;
      outbuf[idx] = res[nt] + x[idx];
    }
  }
}

// ---------------------------------------------------------------------------
// Kernel B: h = out @ W1 (WMMA), accumulate per-column sum / sumsq (h discarded)
// ---------------------------------------------------------------------------
__global__ __launch_bounds__(256) void k_mlp1_stats(
    const float* __restrict__ outbuf, const float* __restrict__ W1,
    float* __restrict__ gsum, float* __restrict__ gsq)
{
  __shared__ float w1s[GN_H * 130];
  __shared__ float ssum[GN_H2];
  __shared__ float ssq[GN_H2];
  const int t = threadIdx.x;
  const int lane = t & 31, wv = t >> 5;
  const int hlf = lane >> 4, l16 = lane & 15;
  const int rowbase = blockIdx.x * 128;

  const float2* w1g = reinterpret_cast<const float2*>(W1);
#pragma unroll
  for (int it = 0; it < 16; ++it) {
    int li = it * 256 + t;
    float2 v = w1g[li];
    int row = li >> 6, col = (li & 63) * 2;
    w1s[row * 130 + col] = v.x; w1s[row * 130 + col + 1] = v.y;
  }
  if (t < GN_H2) { ssum[t] = 0.0f; ssq[t] = 0.0f; }
  __syncthreads();

  const int mrow = rowbase + wv * 16 + l16;
  v8f acc[8] = {};
#pragma unroll
  for (int k0 = 0; k0 < GN_H; k0 += 4) {
    const int ka = k0 + hlf * 2;
    float2 av = *reinterpret_cast<const float2*>(outbuf + (size_t)mrow * GN_H + ka);
    v2f a; a.x = av.x; a.y = av.y;
#pragma unroll
    for (int nt = 0; nt < 8; ++nt) {
      v2f b;
      b.x = w1s[ka * 130 + nt * 16 + l16];
      b.y = w1s[(ka + 1) * 130 + nt * 16 + l16];
      acc[nt] = wmma_f32(a, b, acc[nt]);
    }
  }
#pragma unroll
  for (int nt = 0; nt < 8; ++nt) {
    float s = 0.0f, s2 = 0.0f;
#pragma unroll
    for (int r = 0; r < 8; ++r) {
      const float v = acc[nt][r];
      s += v; s2 += v * v;
    }
    atomicAdd(&ssum[nt * 16 + l16], s);
    atomicAdd(&ssq[nt * 16 + l16], s2);
  }
  __syncthreads();
  if (t < GN_H2) {
    atomicAdd(&gsum[t], ssum[t]);
    atomicAdd(&gsq[t], ssq[t]);
  }
}

// ---------------------------------------------------------------------------
// Kernel C: finalize batch-norm affine params
// ---------------------------------------------------------------------------
__global__ __launch_bounds__(128) void k_bn_finalize(
    const float* __restrict__ gsum, const float* __restrict__ gsq,
    const float* __restrict__ gamma, const float* __restrict__ beta,
    float* __restrict__ scale, float* __restrict__ shift)
{
  const int c = threadIdx.x;
  const float invN = 1.0f / (float)GN_N;
  const float mean = gsum[c] * invN;
  const float var  = gsq[c] * invN - mean * mean;
  const float sc   = gamma[c] * rsqrtf(var + 1e-5f);
  scale[c] = sc;
  shift[c] = beta[c] - mean * sc;
}

__global__ __launch_bounds__(256) void k_zero(float* __restrict__ p, int n)
{
  const int i = blockIdx.x * blockDim.x + threadIdx.x;
  if (i < n) p[i] = 0.0f;
}

// ---------------------------------------------------------------------------
// Kernel D: recompute h = out@W1 tile, normalize+ReLU into LDS (C->A layout
// transpose), then final = h_norm @ W2 via second WMMA stage.
// ---------------------------------------------------------------------------
__global__ __launch_bounds__(256) void k_mlp2(
    const float* __restrict__ outbuf, const float* __restrict__ W1,
    const float* __restrict__ W2, const float* __restrict__ scale,
    const float* __restrict__ shift, float* __restrict__ out)
{
  __shared__ float w1s[GN_H * 130];
  __shared__ float w2s[GN_H2 * 66];
  __shared__ float hs[128 * 130];
  __shared__ float scs[GN_H2];
  __shared__ float shs[GN_H2];
  const int t = threadIdx.x;
  const int lane = t & 31, wv = t >> 5;
  const int hlf = lane >> 4, l16 = lane & 15;
  const int rowbase = blockIdx.x * 128;

  const float2* w1g = reinterpret_cast<const float2*>(W1);
#pragma unroll
  for (int it = 0; it < 16; ++it) {
    int li = it * 256 + t;
    float2 v = w1g[li];
    int row = li >> 6, col = (li & 63) * 2;
    w1s[row * 130 + col] = v.x; w1s[row * 130 + col + 1] = v.y;
  }
  const float2* w2g = reinterpret_cast<const float2*>(W2);
#pragma unroll
  for (int it = 0; it < 16; ++it) {
    int li = it * 256 + t;
    float2 v = w2g[li];
    int row = li >> 5, col = (li & 31) * 2;
    w2s[row * 66 + col] = v.x; w2s[row * 66 + col + 1] = v.y;
  }
  if (t < GN_H2) { scs[t] = scale[t]; shs[t] = shift[t]; }
  __syncthreads();

  // Stage 1: h tile (16 rows per wave), normalized + ReLU, into LDS
  {
    const int mrow = rowbase + wv * 16 + l16;
    v8f acc[8] = {};
#pragma unroll
    for (int k0 = 0; k0 < GN_H; k0 += 4) {
      const int ka = k0 + hlf * 2;
      float2 av = *reinterpret_cast<const float2*>(outbuf + (size_t)mrow * GN_H + ka);
      v2f a; a.x = av.x; a.y = av.y;
#pragma unroll
      for (int nt = 0; nt < 8; ++nt) {
        v2f b;
        b.x = w1s[ka * 130 + nt * 16 + l16];
        b.y = w1s[(ka + 1) * 130 + nt * 16 + l16];
        acc[nt] = wmma_f32(a, b, acc[nt]);
      }
    }
#pragma unroll
    for (int nt = 0; nt < 8; ++nt) {
      const int col = nt * 16 + l16;
#pragma unroll
      for (int r = 0; r < 8; ++r) {
        const int lrow = wv * 16 + r + hlf * 8;
        const float v = acc[nt][r] * scs[col] + shs[col];
        hs[lrow * 130 + col] = fmaxf(v, 0.0f);
      }
    }
  }
  __syncthreads();
  // Stage 2: final = hs @ W2
  {
    v8f acc[4] = {};
    const int lrow = wv * 16 + l16;
#pragma unroll
    for (int k0 = 0; k0 < GN_H2; k0 += 4) {
      const int ka = k0 + hlf * 2;
      v2f a;
      a.x = hs[lrow * 130 + ka];
      a.y = hs[lrow * 130 + ka + 1];
#pragma unroll
      for (int nt = 0; nt < 4; ++nt) {
        v2f b;
        b.x = w2s[ka * 66 + nt * 16 + l16];
        b.y = w2s[(ka + 1) * 66 + nt * 16 + l16];
        acc[nt] = wmma_f32(a, b, acc[nt]);
      }
    }
#pragma unroll
    for (int nt = 0; nt < 4; ++nt)
#pragma unroll
      for (int r = 0; r < 8; ++r) {
        const int orow = rowbase + wv * 16 + r + hlf * 8;
        out[(size_t)orow * GN_H + nt * 16 + l16] = acc[nt][r];
      }
  }
}

// ---------------------------------------------------------------------------
extern "C" void kernel_launch(void* const* d_in, const int* in_sizes, int n_in,
                              void* d_out, int out_size, void* d_ws, size_t ws_size,
                              hipStream_t stream) {
  const float*     x          = (const float*)d_in[0];
  const long long* edge_index = (const long long*)d_in[1];  // [2,E], row0 = src
  const float*     edge_attr  = (const float*)d_in[2];
  const long long* nbr        = (const long long*)d_in[3];
  const float*     W_edge     = (const float*)d_in[4];
  const float*     W1         = (const float*)d_in[5];
  const float*     gamma      = (const float*)d_in[6];
  const float*     beta       = (const float*)d_in[7];
  const float*     W2         = (const float*)d_in[8];
  float* out = (float*)d_out;

  float* ws     = (float*)d_ws;
  float* outbuf = ws;                                // N*64 floats (8 MB)
  float* gsum   = outbuf + (size_t)GN_N * GN_H;      // 128
  float* gsq    = gsum + GN_H2;                      // 128
  float* scale  = gsq + GN_H2;                       // 128
  float* shift  = scale + GN_H2;                     // 128

  k_gnn_fused<<<GN_N / GN_NPB, 32 * GN_NPB, 0, stream>>>(
      x, edge_index, nbr, edge_attr, W_edge, outbuf);
  k_zero<<<1, 256, 0, stream>>>(gsum, 2 * GN_H2);
  k_mlp1_stats<<<GN_N / 128, 256, 0, stream>>>(outbuf, W1, gsum, gsq);
  k_bn_finalize<<<1, 128, 0, stream>>>(gsum, gsq, gamma, beta, scale, shift);
  k_mlp2<<<GN_N / 128, 256, 0, stream>>>(outbuf, W1, W2, scale, shift, out);
}